// GCN_Net_57105885167693
// MI455X (gfx1250) — compile-verified
//
#include <hip/hip_runtime.h>
#include <hip/hip_bf16.h>

#define NODES 50000
#define EDGES 1600000
#define FEAT  256
#define HID   64
#define CLS   40

typedef __attribute__((ext_vector_type(2))) float v2f;
typedef __attribute__((ext_vector_type(8))) float v8f;

// ---------------------------------------------------------------- init / norm
__global__ __launch_bounds__(256) void k_init(float* __restrict__ deg,
                                              float* __restrict__ agg1,
                                              float* __restrict__ agg2) {
  int t = blockIdx.x * blockDim.x + threadIdx.x;
  if (t < NODES) deg[t] = 1.0f;              // self-loop weight (fill_value=1)
  if (t < NODES * HID) agg1[t] = 0.0f;
  if (t < NODES * CLS) agg2[t] = 0.0f;
}

__global__ __launch_bounds__(256) void k_deg(const int* __restrict__ ei,
                                             const float* __restrict__ w,
                                             float* __restrict__ deg) {
  int e = blockIdx.x * blockDim.x + threadIdx.x;
  if (e >= EDGES) return;
  atomicAdd(&deg[ei[EDGES + e]], w[e]);      // in-degree at targets
}

__global__ __launch_bounds__(256) void k_dinv(float* __restrict__ deg) {
  int i = blockIdx.x * blockDim.x + threadIdx.x;
  if (i >= NODES) return;
  float d = deg[i];
  deg[i] = (d > 0.0f) ? rsqrtf(d) : 0.0f;    // deg -> dinv in place
}

// ------------------------------------------------------- fp32 WMMA tile GEMM
// One wave computes a 16x16 C tile with V_WMMA_F32_16X16X4_F32.
// A layout : lane L -> (M = L&15, K = k0 + 2*(L>>4) + {0,1})  (float2 load)
// B layout : VGPR r -> (K = k0 + r + 2*(L>>4), N = L&15)
// C layout : VGPR r -> (M = r + 8*(L>>4),      N = L&15)
// All strides compile-time: addresses = per-lane base + immediate offsets,
// K-loop fully unrolled, no exec-mask churn (OOB columns are clamped, not
// masked, since each C column depends only on its own B column; the store
// guard drops them).
template <int LDA, int LDB, int LDC, int K, int NCOLS, int NTILES>
__global__ __launch_bounds__(256) void k_gemm(const float* __restrict__ A,
                                              const float* __restrict__ B,
                                              float* __restrict__ C,
                                              int totalTiles) {
  int wave = (blockIdx.x * blockDim.x + threadIdx.x) >> 5;
  if (wave >= totalTiles) return;            // wave-uniform: EXEC all-1 inside
  int lane = threadIdx.x & 31;
  int mt = wave / NTILES;
  int nt = wave - mt * NTILES;
  int m0 = mt << 4;
  int kh2 = (lane >> 4) << 1;                // 0 or 2
  int ncol = (nt << 4) + (lane & 15);
  int ncs = ncol < NCOLS ? ncol : NCOLS - 1; // clamp: unused cols never stored
  const float* arow = A + (size_t)(m0 + (lane & 15)) * LDA + kh2;
  const float* bcol = B + (size_t)kh2 * LDB + ncs;

  v8f acc = {};
#pragma unroll
  for (int k0 = 0; k0 < K; k0 += 4) {
    v2f a = *(const v2f*)(arow + k0);        // 8B-aligned (LDA, kh2, k0 even)
    v2f b;
    b.x = bcol[k0 * LDB];
    b.y = bcol[k0 * LDB + LDB];
    // 8 args: (neg_a, A, neg_b, B, c_mod, C, reuse_a, reuse_b)
    acc = __builtin_amdgcn_wmma_f32_16x16x4_f32(false, a, false, b,
                                                (short)0, acc, false, false);
  }
  if (ncol < NCOLS) {
    int mbase = m0 + ((lane >> 4) << 3);
#pragma unroll
    for (int r = 0; r < 8; ++r)
      C[(size_t)(mbase + r) * LDC + ncol] = acc[r];
  }
}

// -------------------------------------------------- edge gather/scale/scatter
// One wave per edge: lanes stride over F features; atomics land in L2
// (agg buffers are <=12.8MB, fully L2-resident on 192MB L2).
template <int F>
__global__ __launch_bounds__(256) void k_agg(const int* __restrict__ ei,
                                             const float* __restrict__ w,
                                             const float* __restrict__ dinv,
                                             const float* __restrict__ h,
                                             float* __restrict__ out) {
  int e = (blockIdx.x * blockDim.x + threadIdx.x) >> 5;
  if (e >= EDGES) return;
  int lane = threadIdx.x & 31;
  int src = ei[e];
  int dst = ei[EDGES + e];
  float coef = dinv[src] * w[e] * dinv[dst];
  const float* hs = h + (size_t)src * F;
  float* od = out + (size_t)dst * F;
#pragma unroll
  for (int f = lane; f < F; f += 32)
    atomicAdd(&od[f], hs[f] * coef);
}

// ------------------------------------- self-loop + bias + ReLU (layer 1 tail)
__global__ __launch_bounds__(256) void k_bias_relu(const float* __restrict__ h1,
                                                   const float* __restrict__ dinv,
                                                   const float* __restrict__ b1,
                                                   float* __restrict__ agg1) {
  int t = blockIdx.x * blockDim.x + threadIdx.x;
  if (t >= NODES * HID) return;
  int i = t >> 6;                            // HID == 64
  int f = t & 63;
  float di = dinv[i];
  float v = agg1[t] + h1[t] * di * di + b1[f];   // self-loop coeff = dinv^2
  agg1[t] = v > 0.0f ? v : 0.0f;
}

// ------------------------- self-loop + bias + log_softmax (layer 2 tail)
__global__ __launch_bounds__(256) void k_final(const float* __restrict__ h2,
                                               const float* __restrict__ agg2,
                                               const float* __restrict__ dinv,
                                               const float* __restrict__ b2,
                                               float* __restrict__ out) {
  int i = blockIdx.x * blockDim.x + threadIdx.x;
  if (i >= NODES) return;
  float di = dinv[i];
  float d2 = di * di;
  const float* hrow = h2 + (size_t)i * CLS;
  const float* arow = agg2 + (size_t)i * CLS;
  float v[CLS];
  float mx = -1e30f;
#pragma unroll
  for (int f = 0; f < CLS; ++f) {
    float t = arow[f] + hrow[f] * d2 + b2[f];
    v[f] = t;
    mx = fmaxf(mx, t);
  }
  float s = 0.0f;
#pragma unroll
  for (int f = 0; f < CLS; ++f) s += expf(v[f] - mx);
  float lse = mx + logf(s);
  float* orow = out + (size_t)i * CLS;
#pragma unroll
  for (int f = 0; f < CLS; ++f) orow[f] = v[f] - lse;
}

// --------------------------------------------------------------------- launch
extern "C" void kernel_launch(void* const* d_in, const int* in_sizes, int n_in,
                              void* d_out, int out_size, void* d_ws, size_t ws_size,
                              hipStream_t stream) {
  (void)in_sizes; (void)n_in; (void)out_size; (void)ws_size;
  const float* x  = (const float*)d_in[0];
  const int*   ei = (const int*)d_in[1];
  const float* ew = (const float*)d_in[2];
  const float* W1 = (const float*)d_in[3];
  const float* b1 = (const float*)d_in[4];
  const float* W2 = (const float*)d_in[5];
  const float* b2 = (const float*)d_in[6];
  float* out = (float*)d_out;

  // workspace layout (floats): dinv | h1 | agg1 | h2 | agg2   (~42 MB total)
  float* ws   = (float*)d_ws;
  float* dinv = ws;                               // 50000 (deg -> dinv)
  float* h1   = ws + (1 << 16);                   // 50000*64
  float* agg1 = h1 + (size_t)NODES * HID;         // 50000*64
  float* h2   = agg1 + (size_t)NODES * HID;       // 50000*40
  float* agg2 = h2 + (size_t)NODES * CLS;         // 50000*40

  // 1) init deg=1 (self loop) and zero aggregation buffers
  k_init<<<(NODES * HID + 255) / 256, 256, 0, stream>>>(dinv, agg1, agg2);
  // 2) degree accumulate, 3) dinv
  k_deg<<<(EDGES + 255) / 256, 256, 0, stream>>>(ei, ew, dinv);
  k_dinv<<<(NODES + 255) / 256, 256, 0, stream>>>(dinv);

  // 4) h1 = x @ W1  (3125 x 4 tiles, K=256, full static unroll: 64 WMMA)
  {
    constexpr int NT = HID / 16;
    int waves = (NODES / 16) * NT;
    k_gemm<FEAT, HID, HID, FEAT, HID, NT>
        <<<(waves * 32 + 255) / 256, 256, 0, stream>>>(x, W1, h1, waves);
  }
  // 5) edge aggregation layer 1 (wave per edge)
  k_agg<HID><<<EDGES / 8, 256, 0, stream>>>(ei, ew, dinv, h1, agg1);
  // 6) + self-loop + b1, ReLU (in place into agg1)
  k_bias_relu<<<(NODES * HID + 255) / 256, 256, 0, stream>>>(h1, dinv, b1, agg1);

  // 7) h2 = relu(h) @ W2  (3125 x 3 tiles, K=64, Ncols=40 guarded: 16 WMMA)
  {
    constexpr int NT = (CLS + 15) / 16;
    int waves = (NODES / 16) * NT;
    k_gemm<HID, CLS, CLS, HID, CLS, NT>
        <<<(waves * 32 + 255) / 256, 256, 0, stream>>>(agg1, W2, h2, waves);
  }
  // 8) edge aggregation layer 2
  k_agg<CLS><<<EDGES / 8, 256, 0, stream>>>(ei, ew, dinv, h2, agg2);
  // 9) + self-loop + b2, log_softmax -> d_out
  k_final<<<(NODES + 255) / 256, 256, 0, stream>>>(h2, agg2, dinv, b2, out);
}